// RiemannianMetric_11553462026875
// MI455X (gfx1250) — compile-verified
//
#include <hip/hip_runtime.h>
#include <hip/hip_bf16.h>

// out[b,i,j] = I[i,j] + (F F^T)[i,j] + s[b]*x[b,i]*x[b,j],  s[b] = 1 - sum_k p_k^2
// B=2048, D=256, out = 512MB f32 -> store-bandwidth bound (~23us at 23.3TB/s).
// Perturbation GEMM done with V_WMMA_F32_16X16X4_F32 (full f32 precision).

#define D_DIM 256
#define B_DIM 2048

typedef __attribute__((ext_vector_type(2))) float v2f;
typedef __attribute__((ext_vector_type(4))) float v4f;
typedef __attribute__((ext_vector_type(8))) float v8f;

// ---------------------------------------------------------------------------
// Kernel A: s[b] = 1 - (sum e_k^2) / (sum e_k)^2, e_k = exp(x_k - max)
// One wave32 per row; 8 elements per lane; shfl_xor reductions.
// ---------------------------------------------------------------------------
__global__ void __launch_bounds__(256) softmax_s_kernel(const float* __restrict__ x,
                                                        float* __restrict__ s) {
    const int wave = threadIdx.x >> 5;
    const int lane = threadIdx.x & 31;
    const int b    = blockIdx.x * 8 + wave;
    const v4f* row4 = (const v4f*)(x + (size_t)b * D_DIM);

    v4f v0 = row4[lane * 2 + 0];
    v4f v1 = row4[lane * 2 + 1];

    float m = fmaxf(fmaxf(fmaxf(v0.x, v0.y), fmaxf(v0.z, v0.w)),
                    fmaxf(fmaxf(v1.x, v1.y), fmaxf(v1.z, v1.w)));
#pragma unroll
    for (int off = 16; off > 0; off >>= 1)
        m = fmaxf(m, __shfl_xor(m, off, 32));

    float Z = 0.0f, E2 = 0.0f;
    float e;
    e = __expf(v0.x - m); Z += e; E2 += e * e;
    e = __expf(v0.y - m); Z += e; E2 += e * e;
    e = __expf(v0.z - m); Z += e; E2 += e * e;
    e = __expf(v0.w - m); Z += e; E2 += e * e;
    e = __expf(v1.x - m); Z += e; E2 += e * e;
    e = __expf(v1.y - m); Z += e; E2 += e * e;
    e = __expf(v1.z - m); Z += e; E2 += e * e;
    e = __expf(v1.w - m); Z += e; E2 += e * e;
#pragma unroll
    for (int off = 16; off > 0; off >>= 1) {
        Z  += __shfl_xor(Z,  off, 32);
        E2 += __shfl_xor(E2, off, 32);
    }
    if (lane == 0) s[b] = 1.0f - E2 / (Z * Z);
}

// ---------------------------------------------------------------------------
// Kernel B: P = F * F^T via V_WMMA_F32_16X16X4_F32.
// One wave per 16x16 output tile; K loop = 64 WMMAs.
// A(16x4) layout: lanes 0-15 = rows M; VGPR0/1 = K{0,1} (lanes<16) or K{2,3}.
// B(4x16) = F^T tile -> identical float2 load pattern with j0 as the row base.
// D(16x16): VGPR v -> M = v (lanes 0-15), M = 8+v (lanes 16-31); N = lane&15.
// ---------------------------------------------------------------------------
__global__ void __launch_bounds__(32) pert_wmma_kernel(const float* __restrict__ mf,
                                                       const int* __restrict__ chartp,
                                                       float* __restrict__ P) {
    const float* F = mf + (size_t)chartp[0] * D_DIM * D_DIM;
    const int lane = threadIdx.x & 31;
    const int i0 = (blockIdx.x >> 4) * 16;
    const int j0 = (blockIdx.x & 15) * 16;
    const int r  = lane & 15;
    const int kh = (lane >> 4) << 1;  // 0 or 2: which K pair this half-wave holds

    const float* arow = F + (size_t)(i0 + r) * D_DIM + kh;
    const float* brow = F + (size_t)(j0 + r) * D_DIM + kh;

    v8f c = {};
#pragma unroll 4
    for (int k = 0; k < D_DIM; k += 4) {
        v2f a = *(const v2f*)(arow + k);
        v2f b = *(const v2f*)(brow + k);
        c = __builtin_amdgcn_wmma_f32_16x16x4_f32(
                /*neg_a=*/false, a, /*neg_b=*/false, b,
                /*c_mod=*/(short)0, c, /*reuse_a=*/false, /*reuse_b=*/false);
    }

    const int mbase = (lane >> 4) * 8;
    const int n = lane & 15;
#pragma unroll
    for (int v = 0; v < 8; ++v)
        P[(size_t)(i0 + mbase + v) * D_DIM + (j0 + n)] = c[v];
}

// ---------------------------------------------------------------------------
// Kernel C: stream 512MB of output. One v4f per thread.
// out[b,i,j..j+3] = P[i,j..] + s[b]*x[b,i]*x[b,j..] (+1 on diagonal).
// Nontemporal stores: keep the write stream out of L2; P/x/s stay cached.
// ---------------------------------------------------------------------------
__global__ void __launch_bounds__(256) fisher_out_kernel(const float* __restrict__ x,
                                                         const float* __restrict__ P,
                                                         const float* __restrict__ s,
                                                         float* __restrict__ out) {
    const size_t gid = (size_t)blockIdx.x * 256 + threadIdx.x;
    const int j4 = (int)(gid & 63);          // v4f index within row of 256
    const int i  = (int)((gid >> 6) & 255);  // output row
    const int b  = (int)(gid >> 14);         // batch

    v4f p4 = ((const v4f*)(P + (size_t)i * D_DIM))[j4];
    v4f x4 = ((const v4f*)(x + (size_t)b * D_DIM))[j4];
    float coef = s[b] * x[(size_t)b * D_DIM + i];

    v4f o;
    o.x = p4.x + coef * x4.x;
    o.y = p4.y + coef * x4.y;
    o.z = p4.z + coef * x4.z;
    o.w = p4.w + coef * x4.w;

    const int d = i - (j4 << 2);  // identity diagonal
    if (d == 0)      o.x += 1.0f;
    else if (d == 1) o.y += 1.0f;
    else if (d == 2) o.z += 1.0f;
    else if (d == 3) o.w += 1.0f;

    __builtin_nontemporal_store(o, (v4f*)out + gid);
}

extern "C" void kernel_launch(void* const* d_in, const int* in_sizes, int n_in,
                              void* d_out, int out_size, void* d_ws, size_t ws_size,
                              hipStream_t stream) {
    const float* x     = (const float*)d_in[0];   // [2048, 256]
    const float* mf    = (const float*)d_in[1];   // [4, 256, 256]
    const int*   chart = (const int*)d_in[2];     // scalar on device
    float* out = (float*)d_out;                   // [2048, 256, 256]

    float* P = (float*)d_ws;                      // 256*256 floats = 256 KB
    float* s = (float*)d_ws + D_DIM * D_DIM;      // 2048 floats

    // s[b]: 2048 rows, 8 waves/block -> 256 blocks
    softmax_s_kernel<<<B_DIM / 8, 256, 0, stream>>>(x, s);

    // P = F F^T: 16x16 tiles -> 256 tiles, one wave each
    pert_wmma_kernel<<<256, 32, 0, stream>>>(mf, chart, P);

    // Output stream: 2048*256*64 v4f elements / 256 threads per block
    const int total_vec4 = B_DIM * D_DIM * (D_DIM / 4);
    fisher_out_kernel<<<total_vec4 / 256, 256, 0, stream>>>(x, P, s, out);
}